// MemoryBank_51780125721167
// MI455X (gfx1250) — compile-verified
//
#include <hip/hip_runtime.h>
#include <math.h>

#define EMA_C          0.99f
#define ONE_MINUS_EMA  0.01f
#define THRESH_C       0.8f
#define EPS_C          1e-8f
#define MSPLIT         16

typedef float __attribute__((ext_vector_type(2))) v2f;
typedef float __attribute__((ext_vector_type(8))) v8f;

// ---------------------------------------------------------------------------
// Phase 1: flag conf > THRESH, exclusive prefix sum, compact (stable order).
// One block of 1024 threads; each thread owns a contiguous item range so the
// compacted order matches the sequential scan order.
// ---------------------------------------------------------------------------
__global__ void mb_scan_compact(const float* __restrict__ conf, int B, int items,
                                int* __restrict__ highSrc, float* __restrict__ highConf,
                                int* __restrict__ numHigh) {
  __shared__ int s[1024];
  const int t = threadIdx.x;
  const int base = t * items;
  int cnt = 0;
  for (int j = 0; j < items; ++j) {
    int i = base + j;
    if (i < B && conf[i] > THRESH_C) ++cnt;
  }
  s[t] = cnt;
  __syncthreads();
  // Hillis-Steele inclusive scan over 1024 thread counts
  for (int off = 1; off < 1024; off <<= 1) {
    int add = (t >= off) ? s[t - off] : 0;
    __syncthreads();
    s[t] += add;
    __syncthreads();
  }
  int rank = s[t] - cnt;  // exclusive
  for (int j = 0; j < items; ++j) {
    int i = base + j;
    if (i < B) {
      float v = conf[i];
      if (v > THRESH_C) { highSrc[rank] = i; highConf[rank] = v; ++rank; }
    }
  }
  if (t == 1023) *numHigh = s[1023];
}

// ---------------------------------------------------------------------------
// Phase 2: per-slot update. Slot p is touched by high items k = k0, k0+M, ...
// (k0 = (p - ptr0) mod M). filled_before_k = min(M, max(filled0, ptr0+k)),
// exists = filled_before_k > p. Apply the (usually length 0/1) chain in order.
// Writes updated bank row into out[p] and masked weight w[p].
// ---------------------------------------------------------------------------
__global__ void mb_update_rows(const float* __restrict__ memF,
                               const float* __restrict__ memC,
                               const float* __restrict__ feat,
                               const int* __restrict__ pPtr0,
                               const int* __restrict__ pFilled0,
                               const int* __restrict__ pNumHigh,
                               const int* __restrict__ highSrc,
                               const float* __restrict__ highConf,
                               float* __restrict__ out,
                               float* __restrict__ w,
                               int M, int C) {
  const int p = blockIdx.x;
  const int t = threadIdx.x;
  const int ptr0 = *pPtr0;
  const int filled0 = *pFilled0;
  const int nH = *pNumHigh;
  const int pm = ((ptr0 % M) + M) % M;
  const int k0 = (((p - pm) % M) + M) % M;

  const float4* __restrict__ src = (const float4*)(memF + (size_t)p * C);
  float4* __restrict__ dst = (float4*)(out + (size_t)p * C);
  const int n4 = C >> 2;

  for (int idx = t; idx < n4; idx += blockDim.x) {
    float4 f = src[idx];
    for (int k = k0; k < nH; k += M) {
      int fb = ptr0 + k;
      if (fb < filled0) fb = filled0;
      if (fb > M) fb = M;
      const bool exists = fb > p;
      const float4* __restrict__ fr = (const float4*)(feat + (size_t)highSrc[k] * C);
      float4 nf = fr[idx];
      if (exists) {
        f.x = EMA_C * f.x + ONE_MINUS_EMA * nf.x;
        f.y = EMA_C * f.y + ONE_MINUS_EMA * nf.y;
        f.z = EMA_C * f.z + ONE_MINUS_EMA * nf.z;
        f.w = EMA_C * f.w + ONE_MINUS_EMA * nf.w;
      } else {
        f = nf;
      }
    }
    dst[idx] = f;
  }

  if (t == 0) {
    float c = memC[p];
    for (int k = k0; k < nH; k += M) {
      int fb = ptr0 + k;
      if (fb < filled0) fb = filled0;
      if (fb > M) fb = M;
      const bool exists = fb > p;
      float nc = highConf[k];
      c = exists ? (EMA_C * c + ONE_MINUS_EMA * nc) : nc;
    }
    int filledF = ptr0 + nH;
    if (filledF < filled0) filledF = filled0;
    if (filledF > M) filledF = M;
    w[p] = (p < filledF) ? c : 0.0f;
  }
}

// ---------------------------------------------------------------------------
// Phase 3: deterministic fixed-order reduction of total weight.
// ---------------------------------------------------------------------------
__global__ void mb_sum_w(const float* __restrict__ w, int M, float* __restrict__ totalW) {
  __shared__ float s[1024];
  const int t = threadIdx.x;
  float acc = 0.0f;
  for (int i = t; i < M; i += 1024) acc += w[i];
  s[t] = acc;
  __syncthreads();
  for (int off = 512; off > 0; off >>= 1) {
    if (t < off) s[t] += s[t + off];
    __syncthreads();
  }
  if (t == 0) *totalW = s[0];
}

// ---------------------------------------------------------------------------
// Phase 4: weighted first/second moments via V_WMMA_F32_16X16X4_F32.
// A (16x4 f32): all 16 rows replicated with w[m0..m0+3]  -> D rows identical.
// B (4x16 f32): tile of the updated bank; second WMMA uses B squared.
// ISA layout (32-bit A 16x4): lanes 0-15 hold K={0,1} in VGPR{0,1},
// lanes 16-31 hold K={2,3}; B mirrors this with N = lane & 15.
// One wave per block; EXEC all-1s through the WMMA loop.
// ---------------------------------------------------------------------------
__global__ void mb_wmma_moments(const float* __restrict__ X,
                                const float* __restrict__ w,
                                float* __restrict__ p1, float* __restrict__ p2,
                                int C, int rowsPerSplit) {
  const int lane = threadIdx.x;           // 0..31
  const int half = lane >> 4;             // 0 or 1
  const int col  = (blockIdx.x << 4) + (lane & 15);
  const int m0b  = blockIdx.y * rowsPerSplit;

  v8f acc1 = {};
  v8f acc2 = {};
  for (int m0 = m0b; m0 < m0b + rowsPerSplit; m0 += 4) {
    const int r0 = m0 + (half << 1);
    const int r1 = r0 + 1;
    v2f a, b, b2;
    a.x = w[r0];
    a.y = w[r1];
    b.x = X[(size_t)r0 * C + col];
    b.y = X[(size_t)r1 * C + col];
    b2.x = b.x * b.x;
    b2.y = b.y * b.y;
    acc1 = __builtin_amdgcn_wmma_f32_16x16x4_f32(false, a, false, b,  (short)0, acc1, false, false);
    acc2 = __builtin_amdgcn_wmma_f32_16x16x4_f32(false, a, false, b2, (short)0, acc2, false, false);
  }
  if (lane < 16) {
    // D VGPR0, lanes 0-15: element (M=0, N=lane); all D rows equal by design.
    p1[(size_t)blockIdx.y * C + col] = acc1[0];
    p2[(size_t)blockIdx.y * C + col] = acc2[0];
  }
}

// ---------------------------------------------------------------------------
// Phase 5: fold the MSPLIT partials (fixed order), emit mean and std rows.
// ---------------------------------------------------------------------------
__global__ void mb_finalize(const float* __restrict__ p1, const float* __restrict__ p2,
                            const float* __restrict__ totalW,
                            float* __restrict__ out, int M, int C) {
  const int c = blockIdx.x * blockDim.x + threadIdx.x;
  if (c >= C) return;
  const float total = *totalW + EPS_C;
  float s1 = 0.0f, s2 = 0.0f;
  for (int s = 0; s < MSPLIT; ++s) {
    s1 += p1[(size_t)s * C + c];
    s2 += p2[(size_t)s * C + c];
  }
  const float mean = s1 / total;
  float var = s2 / total - mean * mean;
  var = fmaxf(var, 0.0f);
  out[(size_t)M * C + c] = mean;
  out[(size_t)(M + 1) * C + c] = sqrtf(var + EPS_C);
}

// ---------------------------------------------------------------------------
extern "C" void kernel_launch(void* const* d_in, const int* in_sizes, int n_in,
                              void* d_out, int out_size, void* d_ws, size_t ws_size,
                              hipStream_t stream) {
  const float* features   = (const float*)d_in[0];
  const float* confidence = (const float*)d_in[1];
  const float* memF       = (const float*)d_in[2];
  const float* memC       = (const float*)d_in[3];
  const int*   ptr0       = (const int*)d_in[4];
  const int*   filled0    = (const int*)d_in[5];
  float* out = (float*)d_out;

  const int B = in_sizes[1];
  const int C = in_sizes[0] / B;
  const int M = in_sizes[3];

  // Workspace carve-up (256B aligned blocks)
  char* ws = (char*)d_ws;
  size_t off = 0;
  auto carve = [&](size_t bytes) -> char* {
    char* p = ws + off;
    off = (off + bytes + 255) & ~(size_t)255;
    return p;
  };
  int*   numHigh  = (int*)carve(sizeof(int));
  int*   highSrc  = (int*)carve((size_t)B * sizeof(int));
  float* highConf = (float*)carve((size_t)B * sizeof(float));
  float* wvec     = (float*)carve((size_t)M * sizeof(float));
  float* totalW   = (float*)carve(sizeof(float));
  float* part1    = (float*)carve((size_t)MSPLIT * C * sizeof(float));
  float* part2    = (float*)carve((size_t)MSPLIT * C * sizeof(float));
  (void)ws_size;

  const int items = (B + 1023) / 1024;
  mb_scan_compact<<<1, 1024, 0, stream>>>(confidence, B, items, highSrc, highConf, numHigh);

  mb_update_rows<<<M, 256, 0, stream>>>(memF, memC, features, ptr0, filled0, numHigh,
                                        highSrc, highConf, out, wvec, M, C);

  mb_sum_w<<<1, 1024, 0, stream>>>(wvec, M, totalW);

  const int rowsPerSplit = M / MSPLIT;                 // 512 for M=8192 (multiple of 4)
  mb_wmma_moments<<<dim3(C / 16, MSPLIT), 32, 0, stream>>>(out, wvec, part1, part2,
                                                           C, rowsPerSplit);

  mb_finalize<<<(C + 255) / 256, 256, 0, stream>>>(part1, part2, totalW, out, M, C);
}